// Model_89704686944627
// MI455X (gfx1250) — compile-verified
//
#include <hip/hip_runtime.h>

// CDNA5 (gfx1250) wave32 WMMA n-body:
//   dist_sq tile  : 1x V_WMMA_F32_16X16X4_F32  (|xi|^2+|xj|^2 in C, -2*xi.xj via A*B)
//   force GEMM    : 4x V_WMMA_F32_16X16X4_F32  (W @ [Xj | 1], accumulated in D over all j-tiles)
//   raw v_rsq_f32 + one Newton step (dist_sq >= eps^2 -> no denorm guard needed)

typedef float v2f __attribute__((ext_vector_type(2)));
typedef float v8f __attribute__((ext_vector_type(8)));

#define EPS2   0.0001f     // SOFTENING^2
#define WROW   20          // padded row stride (floats) for 16x16 LDS staging tile
#define WLDS   336         // per-wave LDS floats (16*WROW=320, padded to 16B multiple)
#define WAVES  2           // waves per block

#if defined(__has_builtin) && __has_builtin(__builtin_amdgcn_rsqf)
#define FAST_RSQ(x) __builtin_amdgcn_rsqf(x)   // bare v_rsq_f32
#else
#define FAST_RSQ(x) rsqrtf(x)
#endif

__global__ __launch_bounds__(WAVES * 32)
void nbody_wmma(const float* __restrict__ pos, const float* __restrict__ mass,
                float* __restrict__ out, int N)
{
    __shared__ float lds[WAVES * WLDS];
    const int lane = threadIdx.x & 31;
    const int wave = threadIdx.x >> 5;
    float* Wt = &lds[wave * WLDS];            // wave-private staging tile

    const int r16 = lane & 15;                // M (A) / N (B,C,D) index for this lane
    const int h   = lane >> 4;                // lane-half selects K pair / M+8
    const int c0  = 2 * h;                    // K base: lanes 0-15 -> K=0,1 ; 16-31 -> K=2,3

    const int numTiles = N >> 4;
    const int iTile = blockIdx.x * WAVES + wave;   // wave-uniform (EXEC stays all-ones)
    if (iTile >= numTiles) return;
    const int iBase = iTile << 4;

    // ---- A matrix for distance GEMM: A[M=r16][K=c0+v] = -2*pos_i[K]; K==3 -> 0 (pad)
    v2f a;
    {
        const float* pI = pos + (size_t)(iBase + r16) * 3;
        a.x = -2.0f * pI[c0];                        // K = 0 or 2 (always a real component)
        a.y = (c0 == 0) ? (-2.0f * pI[1]) : 0.0f;    // K = 1, or K = 3 pad -> 0
    }

    // ---- |x_i|^2 for the 8 rows (M = v + 8h) this lane's C/D elements cover
    float nI[8];
#pragma unroll
    for (int v = 0; v < 8; ++v) {
        const float* p = pos + (size_t)(iBase + v + 8 * h) * 3;
        nI[v] = p[0] * p[0] + p[1] * p[1] + p[2] * p[2];
    }

    v8f acc = {};   // D2[i][c]: c<3 -> sum_j w_ij*x_j[c] ; c==3 -> sum_j w_ij ; c>3 stays 0

    for (int jt = 0; jt < numTiles; ++jt) {
        const int jBase = jt << 4;

        // this lane's column body j = jBase + r16
        const float* pJ = pos + (size_t)(jBase + r16) * 3;
        const float pjx = pJ[0], pjy = pJ[1], pjz = pJ[2];
        const float mj  = mass[jBase + r16];
        const float nje = pjx * pjx + pjy * pjy + pjz * pjz + EPS2;

        // B[K=c0+v][N=r16] = pos_j[K]; K==3 -> 0 (pad)
        v2f b;
        b.x = (c0 == 0) ? pjx : pjz;
        b.y = (c0 == 0) ? pjy : 0.0f;

        // C[M][N] = |x_i|^2 + |x_j|^2 + eps^2
        v8f c;
#pragma unroll
        for (int v = 0; v < 8; ++v) c[v] = nI[v] + nje;

        // D = A*B + C = |xi - xj|^2 + eps^2   (16x16 tile of squared distances)
        v8f d = __builtin_amdgcn_wmma_f32_16x16x4_f32(false, a, false, b,
                                                      (short)0, c, false, false);

        // w_ij = m_j * dist^-3 : raw v_rsq_f32 + one Newton refinement.
        // dist_sq >= eps^2 = 1e-4, so no denorm/overflow range reduction is needed.
        v8f w;
#pragma unroll
        for (int v = 0; v < 8; ++v) {
            const float ds = d[v];
            float t = FAST_RSQ(ds);
            t = t * (1.5f - 0.5f * ds * t * t);
            w[v] = mj * (t * t * t);
        }

        // Stage W (C/D layout) into LDS as Wt[col*WROW + row] so it can be re-read
        // in A-matrix layout. Per lane: element (M = 8h+v, N = r16).
        {
            float4* st = (float4*)&Wt[r16 * WROW + 8 * h];   // 16B aligned (80*r16+32*h)
            st[0] = make_float4(w[0], w[1], w[2], w[3]);
            st[1] = make_float4(w[4], w[5], w[6], w[7]);
        }
        asm volatile("s_wait_dscnt 0" ::: "memory");  // cross-lane LDS RAW inside wave

        // Force GEMM: acc += W @ Pext, Pext[j][c] = (c<3 ? x_j[c] : (c==3 ? 1 : 0)),
        // chained as 4 WMMAs with K=4 each, D-operand accumulation.
#pragma unroll
        for (int k = 0; k < 4; ++k) {
            v2f a2;   // A2[M=r16][K=4k+c0+v] = W[r16][4k+c0+v]
            a2.x = Wt[(4 * k + c0    ) * WROW + r16];
            a2.y = Wt[(4 * k + c0 + 1) * WROW + r16];

            v2f b2;   // B2[K=4k+c0+v][N=r16] = Pext[4k+c0+v][r16]
#pragma unroll
            for (int v = 0; v < 2; ++v) {
                const int jl = 4 * k + c0 + v;
                float bv = 0.0f;
                if (r16 < 3)       bv = pos[(size_t)(jBase + jl) * 3 + r16];
                else if (r16 == 3) bv = 1.0f;
                b2[v] = bv;
            }
            acc = __builtin_amdgcn_wmma_f32_16x16x4_f32(false, a2, false, b2,
                                                        (short)0, acc, false, false);
        }
    }

    // Stage acc (C/D layout) -> Wt[c*WROW + i_local], then finish:
    // F_i[c] = acc[i][c] - acc[i][3] * x_i[c]
    {
        float4* st = (float4*)&Wt[r16 * WROW + 8 * h];
        st[0] = make_float4(acc[0], acc[1], acc[2], acc[3]);
        st[1] = make_float4(acc[4], acc[5], acc[6], acc[7]);
    }
    asm volatile("s_wait_dscnt 0" ::: "memory");

    if (lane < 16) {
        const int i = iBase + lane;
        const float S  = Wt[3 * WROW + lane];
        const float* p = pos + (size_t)i * 3;
        out[(size_t)i * 3 + 0] = Wt[0 * WROW + lane] - S * p[0];
        out[(size_t)i * 3 + 1] = Wt[1 * WROW + lane] - S * p[1];
        out[(size_t)i * 3 + 2] = Wt[2 * WROW + lane] - S * p[2];
    }
}

extern "C" void kernel_launch(void* const* d_in, const int* in_sizes, int n_in,
                              void* d_out, int out_size, void* d_ws, size_t ws_size,
                              hipStream_t stream) {
    const float* pos  = (const float*)d_in[0];   // [N,3] float32
    const float* mass = (const float*)d_in[1];   // [N]   float32
    float* out = (float*)d_out;                  // [N,3] float32

    const int N = in_sizes[0] / 3;               // 8192
    const int numTiles = N >> 4;                 // 512 i-tiles, one wave each
    const int blocks = (numTiles + WAVES - 1) / WAVES;

    hipLaunchKernelGGL(nbody_wmma, dim3(blocks), dim3(WAVES * 32), 0, stream,
                       pos, mass, out, N);
}